// ColorImageLoss_18949395710453
// MI455X (gfx1250) — compile-verified
//
#include <hip/hip_runtime.h>

// CDNA5 / gfx1250: fused gaussian-blur + bilinear-sample + MSE loss.
// Strategy: never materialize the blurred image. Each (point, channel) needs
// only an 8x8 source patch; the 7x7 separable blur for all four bilinear
// corners is one 16x64 * 64x16 f32 matmul per wave, executed as 16 chained
// V_WMMA_F32_16X16X4_F32 with a constant B matrix G[(row,c)][(r,s)] =
// k[row-r]*k[c-s].

typedef __attribute__((ext_vector_type(2))) float v2f;
typedef __attribute__((ext_vector_type(4))) float v4f;
typedef __attribute__((ext_vector_type(8))) float v8f;

#define IMG_H 512
#define IMG_W 512
#define NBATCH 32
#define NPT 64
#define NCH 3
#define NPC (NBATCH * NPT * NCH) /* 6144 point-channels */
#define WAVES_PER_BLOCK 8
#define NBLOCKS 48 /* 48 blocks * 8 waves * 16 pc/wave = 6144 */

__device__ __forceinline__ int mirror512(int t) {
  // jnp.pad 'reflect' (no edge repeat), valid for t in [-3, 515]
  t = (t < 0) ? -t : t;
  t = (t > IMG_W - 1) ? (2 * (IMG_W - 1) - t) : t;
  return t;
}

__global__ __launch_bounds__(256) void ColorImageLoss_main_kernel(
    const float* __restrict__ pred,   // (32, 64, 8)
    const float* __restrict__ img,    // (32, 3, 512, 512)
    float* __restrict__ partial) {    // (NBLOCKS) block partial sums
  __shared__ __align__(16) float stage[WAVES_PER_BLOCK * 256];
  __shared__ float red[256];

  const int t = threadIdx.x;
  const int wave = t >> 5;
  const int L = t & 31;   // lane in wave32
  const int i = L & 15;   // point-channel slot within wave (A-matrix row M)
  const int h = L >> 4;   // K-half for A/B operand layout
  const int g = (blockIdx.x * WAVES_PER_BLOCK + wave) * 16 + i;

  // decode g -> (batch, point, channel)
  const int b = g / (NPT * NCH);
  const int rem = g - b * (NPT * NCH);
  const int pt = rem / NCH;
  const int ch = rem - pt * NCH;

  // prediction: pos = [...,0:2], color = [...,5:8]
  const float* p = pred + (b * NPT + pt) * 8;
  const float px = p[0];
  const float py = p[1];
  const float col = p[5 + ch];

  // sample_bilinear_border coordinate math: x = clip(pos_x*W - 0.5, 0, W-1)
  float x = fminf(fmaxf(px * (float)IMG_W - 0.5f, 0.0f), (float)(IMG_W - 1));
  float y = fminf(fmaxf(py * (float)IMG_H - 0.5f, 0.0f), (float)(IMG_H - 1));
  const float x0f = floorf(x);
  const float y0f = floorf(y);
  const float wx = x - x0f;
  const float wy = y - y0f;
  const int x0 = (int)x0f;
  const int y0 = (int)y0f;

  // reflect-mirrored coordinates for the 8x8 patch around (y0, x0)
  int ym[8], xm[8];
#pragma unroll
  for (int r = 0; r < 8; ++r) ym[r] = mirror512(y0 - 3 + r);
#pragma unroll
  for (int c = 0; c < 8; ++c) xm[c] = mirror512(x0 - 3 + c);

  const float* base = img + (size_t)(b * NCH + ch) * (IMG_H * IMG_W);

  // 7-tap gaussian (sigma=1), normalized
  const float KW[7] = {0.00443304842f, 0.05400558270f, 0.24203622937f,
                       0.39905034602f, 0.24203622937f, 0.05400558270f,
                       0.00443304842f};

  // ---- A operand: patch, 16 chunks of K=4. A 16x4 f32 layout (ISA 7.12.2):
  // VGPR0: lanes0-15 K=0, lanes16-31 K=2; VGPR1: K=1 / K=3; M = lane%16.
  // Lane supplies global K = 4q + 2h + {0,1}; K -> (row = K>>3, col = K&7).
  v2f A[16];
#pragma unroll
  for (int q = 0; q < 16; ++q) {
    const int K0 = 4 * q + 2 * h;
    const int row = K0 >> 3;
    const int c0 = K0 & 7;  // even, <= 6, so c0+1 stays in the same row
    const float* rp = base + ym[row] * IMG_W;
    A[q][0] = rp[xm[c0]];
    A[q][1] = rp[xm[c0 + 1]];
  }

  // ---- B operand: constant combined blur matrix G, sliced per K-chunk.
  // B 4x16 layout assumed symmetric with A/C/D striping:
  // VGPR0: lanes0-15 K=0, lanes16-31 K=2; VGPR1: K=1 / K=3; N = lane%16.
  // Only columns n = 2r+s (n<4) are non-zero: n0=v00, n1=v01, n2=v10, n3=v11.
  const int N = i;
  v2f Bm[16];
#pragma unroll
  for (int q = 0; q < 16; ++q) {
    v2f bv = {0.0f, 0.0f};
    if (N < 4) {
      const int r = N >> 1, s = N & 1;
#pragma unroll
      for (int j = 0; j < 2; ++j) {
        const int K = 4 * q + 2 * h + j;
        const int row = K >> 3, c = K & 7;
        const int jr = row - r, jc = c - s;
        const float wv = (jr >= 0 && jr <= 6) ? KW[jr] : 0.0f;
        const float hv = (jc >= 0 && jc <= 6) ? KW[jc] : 0.0f;
        bv[j] = wv * hv;
      }
    }
    Bm[q] = bv;
  }

  // ---- 16 chained f32 WMMAs: D[i][n] = sum_K patch[i][K] * G[K][n]
  v8f d = {};
#pragma unroll
  for (int q = 0; q < 16; ++q) {
    d = __builtin_amdgcn_wmma_f32_16x16x4_f32(false, A[q], false, Bm[q],
                                              (short)0, d, false, false);
  }

  // ---- stage D through LDS: lane writes (M = r + 8h, N = lane%16)
  float* st = stage + wave * 256;
#pragma unroll
  for (int r = 0; r < 8; ++r) {
    st[(r + 8 * h) * 16 + i] = d[r];
  }
  __syncthreads();

  float err = 0.0f;
  if (L < 16) {
    // lane i collects its 4 corner values: v00, v01, v10, v11
    const v4f v = *(const v4f*)(st + i * 16);
    const float tgt = v[0] * (1.0f - wx) * (1.0f - wy) +
                      v[1] * wx * (1.0f - wy) +
                      v[2] * (1.0f - wx) * wy +
                      v[3] * wx * wy;
    const float dlt = col - tgt;
    err = dlt * dlt;
  }
  red[t] = err;
  __syncthreads();
  if (t == 0) {
    float s = 0.0f;
    for (int k = 0; k < 256; ++k) s += red[k];
    partial[blockIdx.x] = s;
  }
}

__global__ void ColorImageLoss_reduce_kernel(const float* __restrict__ partial,
                                             float* __restrict__ out) {
  float s = 0.0f;
  for (int k = 0; k < NBLOCKS; ++k) s += partial[k];
  out[0] = s * (1.0f / (float)NPC);
}

extern "C" void kernel_launch(void* const* d_in, const int* in_sizes, int n_in,
                              void* d_out, int out_size, void* d_ws,
                              size_t ws_size, hipStream_t stream) {
  const float* pred = (const float*)d_in[0];  // (32,64,8) f32
  const float* img = (const float*)d_in[1];   // (32,3,512,512) f32
  float* partial = (float*)d_ws;              // NBLOCKS floats of scratch

  ColorImageLoss_main_kernel<<<NBLOCKS, 256, 0, stream>>>(pred, img, partial);
  ColorImageLoss_reduce_kernel<<<1, 1, 0, stream>>>(partial, (float*)d_out);
}